// GraphConvLayer_87531433492751
// MI455X (gfx1250) — compile-verified
//
#include <hip/hip_runtime.h>
#include <stdint.h>

typedef __attribute__((ext_vector_type(2))) float    v2f;
typedef __attribute__((ext_vector_type(8))) float    v8f;
typedef __attribute__((ext_vector_type(4))) uint32_t v4u;
typedef __attribute__((ext_vector_type(8))) uint32_t v8u;

#define FDIM 128

// ---------------------------------------------------------------------------
// Kernel 1: zero the workspace (agg + degree) — atomics accumulate, and the
// harness replays the graph, so we must re-zero every launch.
// ---------------------------------------------------------------------------
__global__ void gc_zero_kernel(float* __restrict__ ws, int n) {
    int i = blockIdx.x * blockDim.x + threadIdx.x;
    if (i < n) ws[i] = 0.0f;
}

// ---------------------------------------------------------------------------
// Kernel 2: edge scatter. One wave32 per edge. The edge id is forced into an
// SGPR via readfirstlane so src/dst fetches become scalar loads (KMcnt),
// keeping the vector memory pipe free for the f32 atomics that dominate.
// Each lane: float4 load of x[src] (global_load_b128, coalesced) + 4
// global_atomic_add_f32 into agg[dst]. All traffic is L2-resident.
// ---------------------------------------------------------------------------
__global__ void gc_scatter_kernel(const float* __restrict__ x,
                                  const long long* __restrict__ src,
                                  const long long* __restrict__ dst,
                                  float* __restrict__ agg,
                                  float* __restrict__ deg,
                                  int nE) {
    // wave index as an SGPR value: lane 0's tid >> 5 is the wave id
    int waveInBlk = __builtin_amdgcn_readfirstlane(threadIdx.x) >> 5;
    int edge = blockIdx.x * (blockDim.x >> 5) + waveInBlk;
    int lane = threadIdx.x & 31;
    if (edge >= nE) return;

    long long s = src[edge];   // scalar (s_load_b64)
    long long d = dst[edge];   // scalar (s_load_b64)

    const float4 xv = reinterpret_cast<const float4*>(x + s * FDIM)[lane];
    float* base = agg + d * FDIM + (lane << 2);
    atomicAdd(base + 0, xv.x);
    atomicAdd(base + 1, xv.y);
    atomicAdd(base + 2, xv.z);
    atomicAdd(base + 3, xv.w);
    if (lane == 0) atomicAdd(deg + d, 1.0f);
}

// ---------------------------------------------------------------------------
// Kernel 3: out = relu((agg/deg + x) @ W^T + b) via V_WMMA_F32_16X16X4_F32.
//
// Per block: TDM (tensor_load_to_lds) DMAs all of W (64KB, 2D 128x128 f32)
// into LDS, overlapped with the VALU staging of the 16x128 h-tile. Then 8
// waves each compute one 16x16 output tile with 32 WMMA issues, both A and B
// fragments served from LDS (ds_load_2addr_b64 pairs).
//
// Fragment layouts (CDNA5 ISA 7.12.2, 32-bit operands, wave32):
//   A 16x4 : lane L: M = L%16, VGPR0 = A[M][2*(L/16)], VGPR1 = A[M][2*(L/16)+1]
//   B 4x16 : lane L: N = L%16, VGPR0 = B[2*(L/16)][N], VGPR1 = B[2*(L/16)+1][N]
//   C/D    : VGPR i: M = i + 8*(L/16), N = L%16
// ---------------------------------------------------------------------------
__global__ __launch_bounds__(256)
void gc_gemm_kernel(const float* __restrict__ x,
                    const float* __restrict__ W,
                    const float* __restrict__ bias,
                    const float* __restrict__ agg,
                    const float* __restrict__ deg,
                    float* __restrict__ out) {
    __shared__ float hTile[16 * FDIM];     // 8 KB
    __shared__ float wLds[FDIM * FDIM];    // 64 KB: W row-major

    const int rowBase = blockIdx.x * 16;
    const int t = threadIdx.x;

    // ---- Wave 0 issues the TDM load of W into LDS (TENSORcnt) -------------
    if (t < 32) {
        const uint64_t ga   = (uint64_t)(uintptr_t)W;
        const uint32_t lAdr = (uint32_t)(uintptr_t)wLds;   // LDS byte offset

        // D# group 0 (128b): count=1 | lds_addr | global_addr | type=2
        v4u g0;
        g0.x = 1u;                                   // count=1 (valid), user mode
        g0.y = lAdr;                                 // lds_addr
        g0.z = (uint32_t)(ga & 0xFFFFFFFFu);         // global_addr[31:0]
        g0.w = (uint32_t)((ga >> 32) & 0x01FFFFFFu)  // global_addr[56:32]
               | 0x80000000u;                        // type=2 (bits 127:126)

        // D# group 1 (256b): data_size=4B, 128x128 tensor, 128x128 tile,
        // dim0_stride=128 elements. No multicast / padding / iterate.
        v8u g1;
        g1.s0 = 0x00020000u;        // workgroup_mask=0, data_size=2 (4B)
        g1.s1 = 128u << 16;         // tensor_dim0[15:0]=128 (bits 63:48)
        g1.s2 = 128u << 16;         // tensor_dim0 hi=0, tensor_dim1 lo=128
        g1.s3 = 128u << 16;         // tensor_dim1 hi=0, tile_dim0=128
        g1.s4 = 128u;               // tile_dim1=128, tile_dim2=0
        g1.s5 = 128u;               // tensor_dim0_stride[31:0]=128
        g1.s6 = 0u;                 // dim0_stride hi=0, dim1_stride lo=0
        g1.s7 = 0u;                 // dim1_stride hi=0

        asm volatile("tensor_load_to_lds %0, %1"
                     :: "s"(g0), "s"(g1)
                     : "memory");
    }

    // ---- All threads stage h = agg * (1/max(deg,1)) + x into LDS ----------
    {
        int idx = t * 8;
        int r   = idx >> 7;           // / FDIM
        int c   = idx & (FDIM - 1);
        int row = rowBase + r;
        float inv = 1.0f / fmaxf(deg[row], 1.0f);
        const float* ap = agg + (size_t)row * FDIM + c;
        const float* xp = x   + (size_t)row * FDIM + c;
#pragma unroll
        for (int j = 0; j < 8; ++j)
            hTile[idx + j] = ap[j] * inv + xp[j];
    }

    if (t < 32) {
        __builtin_amdgcn_s_wait_tensorcnt(0);  // W resident in LDS
    }
    __syncthreads();

    // ---- WMMA: one 16x16 tile per wave ------------------------------------
    const int wave    = t >> 5;          // 0..7 -> column tile
    const int lane    = t & 31;
    const int mn      = lane & 15;       // M for A, N for B/D
    const int kb      = (lane >> 4) << 1;
    const int colBase = wave << 4;

    const float* wrow = &wLds[(colBase + mn) * FDIM];   // B[:,n] = W[n,:]
    const float* hrow = &hTile[mn * FDIM];

    v8f acc = {};
#pragma unroll 8
    for (int k0 = 0; k0 < FDIM; k0 += 4) {
        v2f afrag = *reinterpret_cast<const v2f*>(hrow + k0 + kb);
        v2f bfrag = *reinterpret_cast<const v2f*>(wrow + k0 + kb);
        acc = __builtin_amdgcn_wmma_f32_16x16x4_f32(
            /*neg_a=*/false, afrag, /*neg_b=*/false, bfrag,
            /*c_mod=*/(short)0, acc, /*reuse_a=*/false, /*reuse_b=*/false);
    }

    const float bb  = bias[colBase + mn];
    const int   rlo = rowBase + ((lane >> 4) << 3);
#pragma unroll
    for (int i = 0; i < 8; ++i) {
        float v = acc[i] + bb;
        out[(size_t)(rlo + i) * FDIM + colBase + mn] = v > 0.0f ? v : 0.0f;
    }
}

// ---------------------------------------------------------------------------
extern "C" void kernel_launch(void* const* d_in, const int* in_sizes, int n_in,
                              void* d_out, int out_size, void* d_ws, size_t ws_size,
                              hipStream_t stream) {
    const float*     x  = (const float*)d_in[0];
    const long long* ei = (const long long*)d_in[1];   // int64 (2, nE)
    const float*     W  = (const float*)d_in[2];
    const float*     b  = (const float*)d_in[3];

    const int nNodes = in_sizes[0] / FDIM;   // 10000
    const int nE     = in_sizes[1] / 2;      // 640000

    float* agg = (float*)d_ws;                         // nNodes*FDIM floats
    float* deg = agg + (size_t)nNodes * FDIM;          // nNodes floats

    // 1) zero agg + deg
    int nZero = nNodes * FDIM + nNodes;
    gc_zero_kernel<<<(nZero + 255) / 256, 256, 0, stream>>>((float*)d_ws, nZero);

    // 2) scatter-add over edges: one wave32 per edge
    long long nThreads = (long long)nE * 32;
    int nBlocks = (int)((nThreads + 255) / 256);
    gc_scatter_kernel<<<nBlocks, 256, 0, stream>>>(x, ei, ei + nE, agg, deg, nE);

    // 3) TDM-staged W + fused normalize + GEMM(WMMA f32) + bias + ReLU
    gc_gemm_kernel<<<nNodes / 16, 256, 0, stream>>>(x, W, b, agg, deg,
                                                    (float*)d_out);
}